// GTConvFilter_45509473469006
// MI455X (gfx1250) — compile-verified
//
#include <hip/hip_runtime.h>
#include <cstdint>

// GTConv filter: out = (sum_k h[k] * S_k) @ x   (COO scatter-add SpMM)
// N=100000, F=64, K=5, E=1.6M  -> 8M edges total.
//
// Roofline: ~1 GFLOP vs ~96MB HBM edge stream + ~2GB L2 gather + 512M f32
// atomics -> atomic/memory bound, no dense GEMM structure (WMMA inapplicable).
// CDNA5 paths used: async global->LDS staging (double buffered, NT hint,
// s_wait_asynccnt) for the once-read edge stream; native global_atomic_add_f32
// scatter into the L2-resident output; wave32-aware 16-lanes-per-edge layout.

#define TILE 256          // edges per tile == block size
#define MAX_BLOCKS 4096

__device__ __forceinline__ uint32_t lds_offset(const void* p) {
  // gfx1250 flat->LDS mapping uses addr[31:0] as the LDS byte address;
  // the shared aperture selector lives in the high 32 bits.
  return (uint32_t)(uintptr_t)p;
}

__device__ __forceinline__ void async_ld_b32_nt(uint32_t lds_off, const void* gaddr) {
  // GLOBAL_LOAD_ASYNC_TO_LDS_B32 (GV mode): copies one dword from vaddr into LDS at vdst
  // NT temporal hint: edge stream is touched exactly once; keep it from
  // displacing the L2-resident x/out working set.
  asm volatile("global_load_async_to_lds_b32 %0, %1, off th:TH_LOAD_NT"
               :: "v"(lds_off), "v"(gaddr)
               : "memory");
}

__global__ __launch_bounds__(TILE) void gtconv_scatter_kernel(
    const float* __restrict__ x,
    const float* __restrict__ h,
    const uint32_t* __restrict__ vals,
    const uint32_t* __restrict__ rows,
    const uint32_t* __restrict__ cols,
    float* __restrict__ out,
    int total, int E, int ntiles, int K)
{
  __shared__ uint32_t s_val[2][TILE];
  __shared__ uint32_t s_row[2][TILE];
  __shared__ uint32_t s_col[2][TILE];
  __shared__ float    s_h[8];

  const int tid = threadIdx.x;
  if (tid < 8) s_h[tid] = (tid < K) ? h[tid] : 0.0f;

  const int lane = tid & 15;   // which float4 of the 64-float row
  const int grp  = tid >> 4;   // 16 edges processed in parallel per block

  const int stride = gridDim.x;
  int t = blockIdx.x;
  if (t >= ntiles) return;

  // Prologue: stage tile t into buffer 0 (3 async b32 loads per lane).
  {
    int idx = t * TILE + tid; if (idx >= total) idx = total - 1;
    async_ld_b32_nt(lds_offset(&s_val[0][tid]), vals + idx);
    async_ld_b32_nt(lds_offset(&s_row[0][tid]), rows + idx);
    async_ld_b32_nt(lds_offset(&s_col[0][tid]), cols + idx);
  }

  int buf = 0;
  for (; t < ntiles; t += stride) {
    const int tn = t + stride;
    const bool has_next = (tn < ntiles);
    if (has_next) {
      // Prefetch next tile into the other buffer, then wait only for the
      // current tile's 3 loads (async loads complete in issue order).
      const int nb = buf ^ 1;
      int idx = tn * TILE + tid; if (idx >= total) idx = total - 1;
      async_ld_b32_nt(lds_offset(&s_val[nb][tid]), vals + idx);
      async_ld_b32_nt(lds_offset(&s_row[nb][tid]), rows + idx);
      async_ld_b32_nt(lds_offset(&s_col[nb][tid]), cols + idx);
      asm volatile("s_wait_asynccnt 0x3" ::: "memory");
    } else {
      asm volatile("s_wait_asynccnt 0x0" ::: "memory");
    }
    __syncthreads();   // staged tile (all waves) visible; also covers s_h

    const int tbase = t * TILE;
    const int last  = (tbase + TILE - 1 < total) ? (tbase + TILE - 1) : (total - 1);
    const int k0    = tbase / E;          // scalar, once per tile
    const int k1    = last / E;

    if (k0 == k1) {
      // Common case (E % TILE == 0): whole tile belongs to one tap k.
      const float hk = s_h[k0];
#pragma unroll 4
      for (int j = 0; j < TILE; j += 16) {
        const int e   = j + grp;
        const int idx = tbase + e;
        if (idx < total) {
          const uint32_t r = s_row[buf][e];
          const uint32_t c = s_col[buf][e];
          const float    w = hk * __uint_as_float(s_val[buf][e]);

          const float4 xv =
              *(const float4*)(x + (size_t)c * 64u + (unsigned)(lane * 4));
          float* o = out + (size_t)r * 64u + (unsigned)(lane * 4);
          unsafeAtomicAdd(o + 0, w * xv.x);
          unsafeAtomicAdd(o + 1, w * xv.y);
          unsafeAtomicAdd(o + 2, w * xv.z);
          unsafeAtomicAdd(o + 3, w * xv.w);
        }
      }
    } else {
      // Rare generic path: tile crosses a tap boundary.
#pragma unroll 2
      for (int j = 0; j < TILE; j += 16) {
        const int e   = j + grp;
        const int idx = tbase + e;
        if (idx < total) {
          const uint32_t r = s_row[buf][e];
          const uint32_t c = s_col[buf][e];
          const float    w = s_h[idx / E] * __uint_as_float(s_val[buf][e]);

          const float4 xv =
              *(const float4*)(x + (size_t)c * 64u + (unsigned)(lane * 4));
          float* o = out + (size_t)r * 64u + (unsigned)(lane * 4);
          unsafeAtomicAdd(o + 0, w * xv.x);
          unsafeAtomicAdd(o + 1, w * xv.y);
          unsafeAtomicAdd(o + 2, w * xv.z);
          unsafeAtomicAdd(o + 3, w * xv.w);
        }
      }
    }
    __syncthreads();   // everyone done reading buf before it is refilled
    buf ^= 1;
  }
}

extern "C" void kernel_launch(void* const* d_in, const int* in_sizes, int n_in,
                              void* d_out, int out_size, void* d_ws, size_t ws_size,
                              hipStream_t stream) {
  const float*    x    = (const float*)d_in[0];      // [N,64] f32
  const float*    h    = (const float*)d_in[1];      // [K]    f32
  const uint32_t* vals = (const uint32_t*)d_in[2];   // [K*E]  f32 bits
  const uint32_t* rows = (const uint32_t*)d_in[3];   // [K*E]  i32
  const uint32_t* cols = (const uint32_t*)d_in[4];   // [K*E]  i32
  float* out = (float*)d_out;                        // [N,64] f32

  const int K     = in_sizes[1];
  const int total = in_sizes[2];          // K*E edges
  const int E     = total / (K > 0 ? K : 1);
  const int ntiles = (total + TILE - 1) / TILE;
  int grid = ntiles < MAX_BLOCKS ? ntiles : MAX_BLOCKS;
  if (grid < 1) grid = 1;

  hipMemsetAsync(d_out, 0, (size_t)out_size * sizeof(float), stream);
  gtconv_scatter_kernel<<<grid, TILE, 0, stream>>>(x, h, vals, rows, cols, out,
                                                   total, E, ntiles, K);
}